// VectorQuantizer_72619307040978
// MI455X (gfx1250) — compile-verified
//
#include <hip/hip_runtime.h>

typedef __attribute__((ext_vector_type(2))) float v2f;
typedef __attribute__((ext_vector_type(4))) float v4f;
typedef __attribute__((ext_vector_type(8))) float v8f;
typedef unsigned int u32;
typedef __attribute__((ext_vector_type(4))) unsigned int u32x4;
typedef __attribute__((ext_vector_type(4))) int i32x4;
typedef __attribute__((ext_vector_type(8))) int i32x8;

#define N_E        1024
#define E_DIM      64
#define BATCH      131072
#define BETA       0.25f

#define ROWS_PER_BLOCK 64    // 4 waves * 16 rows
#define WAVES          4
#define CHUNK          64    // codebook rows per LDS chunk (double buffered)
#define NCHUNK         (N_E / CHUNK)
#define LDS_STRIDE     68    // 64 + 4 pad words -> conflict-free strided v2f loads

#if __has_builtin(__builtin_amdgcn_tensor_load_to_lds)
#define HAVE_TDM 1
#endif

// ---------------------------------------------------------------------------
// TDM stage: DMA a 16x64 f32 tile (row-major, stride 64) into LDS with
// +4-dword row padding (LDS row stride = 68 dwords). One descriptor per wave.
//   D# group0: count=1 | lds_addr | global_addr | type=2
//   D# group1: data_size=4B, pad_enable, pad_interval=5 (64 dwords),
//              pad_amount=3 (4 dwords), tensor_dim0=64, tensor_dim1=16,
//              tile_dim0=64, tile_dim1=16, tensor_dim0_stride=64
// ---------------------------------------------------------------------------
__device__ __forceinline__ void stage_rows16(const float* __restrict__ gsrc,
                                             float* __restrict__ ldst, int lane) {
#ifdef HAVE_TDM
  // Generic LDS address = {SHARED_BASE[63:32], lds_offset[31:0]} (ISA 10.2)
  unsigned lbyte = (unsigned)(unsigned long long)(void*)ldst;
  unsigned long long ga = (unsigned long long)gsrc;
  u32x4 g0;
  g0.x = 1u;                                        // count=1, user mode
  g0.y = lbyte;                                     // lds_addr
  g0.z = (u32)ga;                                   // global_addr[31:0]
  g0.w = (u32)((ga >> 32) & 0x01FFFFFFu) | (2u << 30);  // global_addr[56:32] | type=2
  i32x8 g1;
  g1[0] = (int)((2u << 16) | (1u << 20) | (5u << 22) | (3u << 25));
  g1[1] = (int)(64u << 16);   // tensor_dim0 = 64
  g1[2] = (int)(16u << 16);   // tensor_dim0 hi=0 | tensor_dim1 = 16
  g1[3] = (int)(64u << 16);   // tensor_dim1 hi=0 | tile_dim0 = 64
  g1[4] = 16;                 // tile_dim1 = 16 | tile_dim2 = 0
  g1[5] = 64;                 // tensor_dim0_stride = 64
  g1[6] = 0;
  g1[7] = 0;
  i32x4 zz = {0, 0, 0, 0};
#if defined(__clang_major__) && (__clang_major__ >= 23)
  i32x8 zz8 = {0, 0, 0, 0, 0, 0, 0, 0};
  __builtin_amdgcn_tensor_load_to_lds(g0, g1, zz, zz, zz8, 0);
#else
  __builtin_amdgcn_tensor_load_to_lds(g0, g1, zz, zz, 0);
#endif
  (void)lane;
#else
  // Fallback: synchronous cooperative copy (2 lanes per row, 8 x float4 each)
  int row = lane >> 1, half = lane & 1;
  const v4f* src = (const v4f*)(gsrc + (size_t)row * E_DIM + half * 32);
  float* dst = ldst + row * LDS_STRIDE + half * 32;
#pragma unroll
  for (int j = 0; j < 8; ++j) *(v4f*)(dst + j * 4) = src[j];
#endif
}

__device__ __forceinline__ void stage_wait_le1() {
#ifdef HAVE_TDM
  __builtin_amdgcn_s_wait_tensorcnt(1);
#endif
}
__device__ __forceinline__ void stage_wait_le0() {
#ifdef HAVE_TDM
  __builtin_amdgcn_s_wait_tensorcnt(0);
#endif
}

// ---------------------------------------------------------------------------
// Kernel 1: e2[c] = sum_d embedding[c][d]^2   (1024 values into d_ws)
// ---------------------------------------------------------------------------
__global__ __launch_bounds__(256) void vq_e2_kernel(const float* __restrict__ emb,
                                                    float* __restrict__ e2) {
  int c = blockIdx.x * blockDim.x + threadIdx.x;
  if (c < N_E) {
    const v4f* p = (const v4f*)(emb + (size_t)c * E_DIM);
    float s = 0.f;
#pragma unroll
    for (int j = 0; j < E_DIM / 4; ++j) {
      v4f q = p[j];
      s += q.x * q.x + q.y * q.y + q.z * q.z + q.w * q.w;
    }
    e2[c] = s;
  }
}

// ---------------------------------------------------------------------------
// Kernel 2: nearest-code search: V_WMMA_F32_16X16X4_F32 over TDM-staged,
// double-buffered codebook chunks; gather z_q; per-block loss partial.
// ---------------------------------------------------------------------------
__global__ __launch_bounds__(128) void vq_main_kernel(const float* __restrict__ z,
                                                      const float* __restrict__ emb,
                                                      const float* __restrict__ e2g,
                                                      float* __restrict__ zq_out,
                                                      float* __restrict__ blk_loss) {
  __shared__ float lds_z[ROWS_PER_BLOCK * LDS_STRIDE];   // 17408 B
  __shared__ float lds_b[2][CHUNK * LDS_STRIDE];         // 34816 B (double buffer)
  __shared__ float lds_e2[N_E];                          // 4096 B
  __shared__ int   lds_idx[ROWS_PER_BLOCK];
  __shared__ float lds_loss[WAVES];

  const int tid   = threadIdx.x;
  const int wave  = tid >> 5;
  const int lane  = tid & 31;
  const int nlo   = lane & 15;            // M (A) / N (B,C) sub-index
  const int khalf = (lane >> 4) << 1;     // K pair: 0 or 2
  const int m0    = blockIdx.x * ROWS_PER_BLOCK;

  // ---- Prologue: TDM-stage this wave's z quarter + chunk0 quarter; e2 table ----
  stage_rows16(z + (size_t)(m0 + wave * 16) * E_DIM,
               &lds_z[wave * 16 * LDS_STRIDE], lane);
  stage_rows16(emb + (size_t)(wave * 16) * E_DIM,
               &lds_b[0][wave * 16 * LDS_STRIDE], lane);
  for (int k = tid; k < N_E; k += 128) lds_e2[k] = e2g[k];
  stage_wait_le0();
  __syncthreads();

  // ---- Preload A fragments (16x4 f32 layout: lane L holds row M=L%16,
  //      K = {0,1} (L<16) or {2,3} (L>=16)) ----
  v2f a[16];
#pragma unroll
  for (int ks = 0; ks < 16; ++ks)
    a[ks] = *(const v2f*)&lds_z[(wave * 16 + nlo) * LDS_STRIDE + ks * 4 + khalf];

  float minval[8];
  int   minidx[8];
#pragma unroll
  for (int v = 0; v < 8; ++v) { minval[v] = 3.4e38f; minidx[v] = 0; }

  // ---- Pipelined sweep: issue TDM for chunk c+1, wait chunk c, compute c ----
  for (int chunk = 0; chunk < NCHUNK; ++chunk) {
    if (chunk + 1 < NCHUNK) {
      stage_rows16(emb + ((size_t)(chunk + 1) * CHUNK + wave * 16) * E_DIM,
                   &lds_b[(chunk + 1) & 1][wave * 16 * LDS_STRIDE], lane);
      stage_wait_le1();                   // chunk's own quarter is done
    } else {
      stage_wait_le0();
    }
    __syncthreads();                      // all 4 quarters of buf[chunk&1] ready

    const float* bbuf = lds_b[chunk & 1];
#pragma unroll
    for (int tile = 0; tile < CHUNK / 16; ++tile) {
      v8f c = {};
#pragma unroll
      for (int ks = 0; ks < 16; ++ks) {
        // B fragment (4x16 f32): lane L holds col N=L%16, rows K={0,1}/{2,3}
        v2f b = *(const v2f*)&bbuf[(tile * 16 + nlo) * LDS_STRIDE + ks * 4 + khalf];
        c = __builtin_amdgcn_wmma_f32_16x16x4_f32(false, a[ks], false, b,
                                                  (short)0, c, false, false);
      }
      // score[m][n] = ||e||^2 - 2 z.e  (||z||^2 constant per row -> dropped)
      float e2v  = lds_e2[chunk * CHUNK + tile * 16 + nlo];
      int   code = chunk * CHUNK + tile * 16 + nlo;
#pragma unroll
      for (int v = 0; v < 8; ++v) {
        float s   = fmaf(-2.0f, c[v], e2v);
        bool take = s < minval[v];
        minval[v] = take ? s : minval[v];
        minidx[v] = take ? code : minidx[v];
      }
    }
    __syncthreads();                      // everyone done reading buf[chunk&1]
  }

  // ---- Argmin across the 16 lanes of each half (C layout: lanes 0-15 -> M=v,
  //      lanes 16-31 -> M=v+8). Tiebreak: lowest index (matches jnp.argmin). ----
#pragma unroll
  for (int v = 0; v < 8; ++v) {
    float mv = minval[v];
    int   mi = minidx[v];
#pragma unroll
    for (int off = 1; off < 16; off <<= 1) {
      float ov = __shfl_xor(mv, off, 32);
      int   oi = __shfl_xor(mi, off, 32);
      bool take = (ov < mv) || (ov == mv && oi < mi);
      mv = take ? ov : mv;
      mi = take ? oi : mi;
    }
    minval[v] = mv;
    minidx[v] = mi;
  }
  if (lane == 0 || lane == 16) {
    int mbase = wave * 16 + ((lane >> 4) << 3);
#pragma unroll
    for (int v = 0; v < 8; ++v) lds_idx[mbase + v] = minidx[v];
  }
  __syncthreads();

  // ---- Gather z_q, write out, and accumulate sum (z - z_q)^2 ----
  float acc = 0.f;
#pragma unroll
  for (int m = 0; m < 16; ++m) {
    int row = wave * 16 + m;
    int idx = lds_idx[row];
    v2f zq  = *(const v2f*)(emb + (size_t)idx * E_DIM + lane * 2);
    v2f zv  = *(const v2f*)&lds_z[row * LDS_STRIDE + lane * 2];
    float dx = zv.x - zq.x, dy = zv.y - zq.y;
    acc += dx * dx + dy * dy;
    *(v2f*)(zq_out + (size_t)(m0 + row) * E_DIM + lane * 2) = zq;
  }
#pragma unroll
  for (int off = 1; off < 32; off <<= 1) acc += __shfl_xor(acc, off, 32);
  if (lane == 0) lds_loss[wave] = acc;
  __syncthreads();
  if (tid == 0) {
    float s = 0.f;
#pragma unroll
    for (int w = 0; w < WAVES; ++w) s += lds_loss[w];
    blk_loss[blockIdx.x] = s;
  }
}

// ---------------------------------------------------------------------------
// Kernel 3: deterministic fixed-order reduction of block partials -> out[0]
// ---------------------------------------------------------------------------
__global__ __launch_bounds__(256) void vq_loss_reduce(const float* __restrict__ blk_loss,
                                                      float* __restrict__ out, int nblk) {
  __shared__ float sm[256];
  int tid = threadIdx.x;
  float s = 0.f;
  for (int i = tid; i < nblk; i += 256) s += blk_loss[i];
  sm[tid] = s;
  __syncthreads();
  for (int off = 128; off > 0; off >>= 1) {
    if (tid < off) sm[tid] += sm[tid + off];
    __syncthreads();
  }
  if (tid == 0) out[0] = (1.0f + BETA) * sm[0];  // embed_loss + beta*commit_loss
}

// ---------------------------------------------------------------------------
extern "C" void kernel_launch(void* const* d_in, const int* in_sizes, int n_in,
                              void* d_out, int out_size, void* d_ws, size_t ws_size,
                              hipStream_t stream) {
  const float* z   = (const float*)d_in[0];   // (131072, 64) f32
  const float* emb = (const float*)d_in[1];   // (1024, 64) f32
  float* out = (float*)d_out;                 // [loss] ++ z_q flat

  float* ws_e2   = (float*)d_ws;              // 1024 floats
  float* ws_loss = ws_e2 + N_E;               // 2048 floats (one per block)
  const int nblk = BATCH / ROWS_PER_BLOCK;    // 2048

  vq_e2_kernel<<<N_E / 256, 256, 0, stream>>>(emb, ws_e2);
  vq_main_kernel<<<nblk, 128, 0, stream>>>(z, emb, ws_e2, out + 1, ws_loss);
  vq_loss_reduce<<<1, 256, 0, stream>>>(ws_loss, out, nblk);
}